// TopologicalPentachoronLoss_84052509983124
// MI455X (gfx1250) — compile-verified
//
#include <hip/hip_runtime.h>
#include <math.h>

typedef float v2f __attribute__((ext_vector_type(2)));
typedef float v8f __attribute__((ext_vector_type(8)));

// ---------------- workspace layout (floats) ----------------
// [0, B*15)        : per-batch 15-float slots, 10 pair distances written
// [B*15, B*16)     : per-batch volume loss
// after that (uints): meta[0]=min bits, meta[1]=max bits, meta[2..10]=counts[9],
//                     meta[11] = final scalar (as float bits)

__device__ inline unsigned int f2u(float f) { return __float_as_uint(f); }
__device__ inline float u2f(unsigned int u) { return __uint_as_float(u); }

// 6x6 determinant, Gaussian elimination with partial pivoting.
__device__ float det6(float M[6][6]) {
    float det = 1.0f;
    for (int c = 0; c < 6; ++c) {
        int p = c;
        float mv = fabsf(M[c][c]);
        for (int r = c + 1; r < 6; ++r) {
            float a = fabsf(M[r][c]);
            if (a > mv) { mv = a; p = r; }
        }
        if (mv == 0.0f) return 0.0f;
        if (p != c) {
            for (int j = c; j < 6; ++j) {
                float t = M[c][j]; M[c][j] = M[p][j]; M[p][j] = t;
            }
            det = -det;
        }
        float pv = M[c][c];
        det *= pv;
        for (int r = c + 1; r < 6; ++r) {
            float f = M[r][c] / pv;
            for (int j = c; j < 6; ++j) M[r][j] -= f * M[c][j];
        }
    }
    return det;
}

__global__ void k_init(unsigned int* meta) {
    int t = threadIdx.x;
    if (t == 0) meta[0] = 0x7F800000u;        // +inf (min positive dist)
    if (t == 1) meta[1] = 0u;                 // max dist (>=0)
    if (t >= 2 && t < 11) meta[t] = 0u;       // counts[9]
}

// One wave computes Gram for 3 batches via V_WMMA_F32_16X16X4_F32.
// Rows 0..14 = vertices of batches b0,b0+1,b0+2 ; row 15 = pad (clamped).
__global__ void __launch_bounds__(256)
k_gram(const float* __restrict__ verts, float* __restrict__ wsd,
       float* __restrict__ wsvol, unsigned int* __restrict__ meta, int B) {
    __shared__ float G[8][16 * 17];           // per-wave 16x16 tile (padded)
    __shared__ unsigned int bmin, bmax;

    const int tid  = threadIdx.x;
    const int wave = tid >> 5;
    const int lane = tid & 31;
    if (tid == 0) { bmin = 0x7F800000u; bmax = 0u; }

    const long wg   = (long)blockIdx.x * 8 + wave;  // global wave id
    const long b0   = wg * 3;                        // first batch of this wave
    const int  r    = lane & 15;                     // tile row (M) / col (N)
    const int  half = lane >> 4;                     // K-half select

    long grow = b0 * 5 + r;                          // global vertex row
    const long maxrow = (long)B * 5 - 1;
    if (grow > maxrow) grow = maxrow;                // clamp: keep EXEC all-ones
    const float* p = verts + grow * 512 + 2 * half;

    v8f acc = {};
    #pragma unroll 8
    for (int k = 0; k < 512; k += 4) {
        // A 16x4 f32 layout: lane<16 holds (k,k+1), lane>=16 holds (k+2,k+3).
        // B = A^T has the *same* per-lane layout, so pass the same regs twice.
        v2f a = *(const v2f*)(p + k);
        acc = __builtin_amdgcn_wmma_f32_16x16x4_f32(
            false, a, false, a, (short)0, acc, false, false);
    }

    __syncthreads();                                  // bmin/bmax init visible

    // Scatter accumulator to LDS: VGPR i, lane -> D[i + 8*half][r]
    float* g = &G[wave][0];
    #pragma unroll
    for (int i = 0; i < 8; ++i) g[(i + 8 * half) * 17 + r] = acc[i];
    __syncthreads();

    if (lane < 3) {
        const long b = b0 + lane;
        if (b < B) {
            const int o = 5 * lane;                   // diag block origin
            float sq[5];
            #pragma unroll
            for (int i = 0; i < 5; ++i) sq[i] = g[(o + i) * 17 + (o + i)];

            float dm[5][5];
            float minp = u2f(0x7F800000u);
            float maxd = 0.0f;
            int t = 0;
            #pragma unroll
            for (int i = 0; i < 5; ++i) {
                dm[i][i] = 0.0f;
                for (int j = i + 1; j < 5; ++j) {
                    float gij = g[(o + i) * 17 + (o + j)];
                    float d2  = fmaxf(sq[i] + sq[j] - 2.0f * gij, 0.0f);
                    float dd  = (d2 > 0.0f) ? sqrtf(d2) : 0.0f;
                    dm[i][j] = dd; dm[j][i] = dd;
                    wsd[b * 15 + t] = dd; ++t;
                    if (dd > 0.0f) minp = fminf(minp, dd);
                    maxd = fmaxf(maxd, dd);
                }
            }

            // Cayley-Menger 6x6
            float CM[6][6];
            CM[0][0] = 0.0f;
            #pragma unroll
            for (int i = 0; i < 5; ++i) { CM[0][i + 1] = 1.0f; CM[i + 1][0] = 1.0f; }
            #pragma unroll
            for (int i = 0; i < 5; ++i)
                for (int j = 0; j < 5; ++j)
                    CM[i + 1][j + 1] = dm[i][j] * dm[i][j];

            float det = det6(CM);
            float vol = sqrtf(fabsf(det) / 384.0f);   // 16*24
            wsvol[b] = fmaxf(1e-6f - vol, 0.0f);

            atomicMin(&bmin, f2u(minp));              // bits order == float order (>=0)
            atomicMax(&bmax, f2u(maxd));
        }
    }
    __syncthreads();
    if (tid == 0) {
        atomicMin(&meta[0], bmin);
        atomicMax(&meta[1], bmax);
    }
}

// Count pairs with d < eps_k for k = 1..9 (eps_0 = 0 contributes nothing).
__global__ void __launch_bounds__(256)
k_pers(const float* __restrict__ wsd, unsigned int* __restrict__ meta, int B) {
    __shared__ unsigned int lc[9];
    const int tid = threadIdx.x;
    if (tid < 9) lc[tid] = 0u;
    __syncthreads();

    const long b = (long)blockIdx.x * blockDim.x + tid;
    const float maxd = u2f(meta[1]);
    if (b < B) {
        unsigned int c[9] = {0, 0, 0, 0, 0, 0, 0, 0, 0};
        #pragma unroll
        for (int t = 0; t < 15; ++t) {
            float dd = wsd[b * 15 + t];
            #pragma unroll
            for (int k = 1; k <= 9; ++k) {
                float eps = maxd * ((float)k / 9.0f);
                c[k - 1] += (dd < eps) ? 1u : 0u;
            }
        }
        #pragma unroll
        for (int k = 0; k < 9; ++k)
            if (c[k]) atomicAdd(&lc[k], c[k]);
    }
    __syncthreads();
    if (tid < 9 && lc[tid]) atomicAdd(&meta[2 + tid], lc[tid]);
}

__global__ void k_final(unsigned int* meta, int B) {
    if (threadIdx.x != 0) return;
    const float minp = u2f(meta[0]);
    const float maxd = u2f(meta[1]);
    const float conn = fmaxf(0.1f - minp, 0.0f) + fmaxf(maxd - 10.0f, 0.0f);

    float pers = 0.0f;
    for (int k = 1; k <= 9; ++k) {
        float eps  = maxd * ((float)k / 9.0f);
        float diag = (eps > 0.0f) ? 5.0f * (float)B : 0.0f;   // diagonal adj hits
        float n    = 0.5f * diag + (float)meta[2 + (k - 1)];  // n_edges
        pers += fmaxf(n - 5.0f, 0.0f);
    }
    pers *= 0.1f;                                             // /10 epsilons
    meta[11] = f2u(conn + 0.1f * pers);
}

__global__ void __launch_bounds__(256)
k_out(const float* __restrict__ wsvol, const unsigned int* __restrict__ meta,
      float* __restrict__ out, int B) {
    const long b = (long)blockIdx.x * blockDim.x + threadIdx.x;
    if (b < B) out[b] = wsvol[b] + u2f(meta[11]);
}

extern "C" void kernel_launch(void* const* d_in, const int* in_sizes, int n_in,
                              void* d_out, int out_size, void* d_ws, size_t ws_size,
                              hipStream_t stream) {
    (void)n_in; (void)ws_size;
    const float* verts = (const float*)d_in[0];
    float* out = (float*)d_out;
    const int B = in_sizes[0] / (5 * 512);
    if (B <= 0) return;

    float* wsf = (float*)d_ws;
    float* wsd = wsf;                              // B*15 distances
    float* wsvol = wsf + (size_t)B * 15;           // B volume losses
    unsigned int* meta = (unsigned int*)(wsvol + B);

    k_init<<<1, 32, 0, stream>>>(meta);

    const int wavesTotal = (B + 2) / 3;            // 3 batches per wave
    const int gramBlocks = (wavesTotal + 7) / 8;   // 8 waves per block
    k_gram<<<gramBlocks, 256, 0, stream>>>(verts, wsd, wsvol, meta, B);

    const int persBlocks = (B + 255) / 256;
    k_pers<<<persBlocks, 256, 0, stream>>>(wsd, meta, B);

    k_final<<<1, 1, 0, stream>>>(meta, B);

    k_out<<<persBlocks, 256, 0, stream>>>(wsvol, meta, out, B);
}